// ModelNew_14061722927798
// MI455X (gfx1250) — compile-verified
//
#include <hip/hip_runtime.h>
#include <hip/hip_bf16.h>

typedef __attribute__((ext_vector_type(2))) float v2f;
typedef __attribute__((ext_vector_type(4))) float v4f;
typedef __attribute__((ext_vector_type(8))) float v8f;

#define BATCH   128
#define CIN     8
#define HIN     128
#define WIN     128
#define COUT    64
#define HOUT    126
#define WOUT    126
#define M_TOT   (HOUT * WOUT)      // 15876 spatial outputs per image
#define KTOT    72                  // 8 * 3 * 3
#define KSTEPS  18                  // 72 / 4
#define GROUPS  16
#define POOLED  31
#define GN_EPS  1e-5f
#define PLANE   (HIN * WIN)         // 16384

// K is permuted tap-major: k_hat = tap*8 + ci  (tap = kh*3+kw).
// For step s (K-chunk of 4) and lane-half hi:
//   tap = s>>1 (uniform per step), ci = 4*(s&1) + 2*hi + j.
// The 2*hi part folds into the per-lane base pointer; everything else is a
// compile-time immediate offset -> global_load_b32/b96 with offset:imm.
__device__ __forceinline__ constexpr int offA(int s, int j) {
    const int tap = s >> 1;
    return (4 * (s & 1) + j) * PLANE + (tap / 3) * WIN + (tap % 3);
}

// ---------------------------------------------------------------------------
// Kernel 1: implicit-GEMM conv via V_WMMA_F32_16X16X4_F32 + group stats
//   grid = (63, 128), block = 256 = 8 waves.
//   Each wave: M = 32 rows (two 16-row WMMA sub-tiles) x N = 64, K = 72.
//   B fragments (LDS) are shared by both sub-tiles and double-buffered.
// ---------------------------------------------------------------------------

__device__ __forceinline__ void epilogue_tile(
    float* __restrict__ convout, size_t plane, int mbase, v8f acc,
    float bn, float* __restrict__ sgrp, int g)
{
    acc = acc + bn;  // channel (=N) is lane-uniform in the C layout
    float s1 = 0.0f, s2 = 0.0f;
    if (mbase + 7 < M_TOT) {
        v4f lo4 = {acc[0], acc[1], acc[2], acc[3]};
        v4f hi4 = {acc[4], acc[5], acc[6], acc[7]};
        __builtin_nontemporal_store(lo4, (v4f*)(convout + plane + mbase));
        __builtin_nontemporal_store(hi4, (v4f*)(convout + plane + mbase + 4));
#pragma unroll
        for (int j = 0; j < 8; ++j) { float v = acc[j]; s1 += v; s2 += v * v; }
    } else {
#pragma unroll
        for (int j = 0; j < 8; ++j) {
            if (mbase + j < M_TOT) {
                float v = acc[j];
                convout[plane + mbase + j] = v;
                s1 += v; s2 += v * v;
            }
        }
    }
    atomicAdd(&sgrp[2 * g],     s1);   // ds_add_f32
    atomicAdd(&sgrp[2 * g + 1], s2);
}

__global__ __launch_bounds__(256) void conv_stats_kernel(
    const float* __restrict__ x,        // [128, 8, 128, 128]
    const float* __restrict__ w,        // [64, 8, 3, 3] == [64][72]
    const float* __restrict__ bias,     // [64]
    float* __restrict__ convout,        // [128, 64, 15876]
    float* __restrict__ stats)          // [128, 16, 2] (sum, sumsq)
{
    // B fragments for all 18 k-steps x 4 n-tiles, WMMA lane layout,
    // tap-major K permutation baked in.
    __shared__ float packedB[KSTEPS * 4 * 32 * 2];   // 18 KB
    __shared__ float sgrp[GROUPS * 2];

    const int tid = threadIdx.x;
    const int b   = blockIdx.y;

    for (int i = tid; i < KSTEPS * 4 * 32; i += 256) {
        int l    = i & 31;
        int t    = (i >> 5) & 3;
        int s    = i >> 7;
        int n    = 16 * t + (l & 15);
        int khat = 4 * s + 2 * (l >> 4);       // permuted K index (j = 0)
        int ci0  = khat & 7, tap0 = khat >> 3;
        int ci1  = (khat + 1) & 7, tap1 = (khat + 1) >> 3;
        packedB[2 * i]     = w[n * KTOT + ci0 * 9 + tap0];
        packedB[2 * i + 1] = w[n * KTOT + ci1 * 9 + tap1];
    }
    if (tid < GROUPS * 2) sgrp[tid] = 0.0f;
    __syncthreads();

    const int lane = tid & 31;
    const int wave = tid >> 5;
    const int hi   = lane >> 4;
    const int lo   = lane & 15;
    const int m0   = blockIdx.x * 256 + wave * 32;   // wave owns rows m0..m0+31

    // Per-lane A base pointers for the two 16-row sub-tiles (clamped rows
    // keep EXEC all-ones; invalid rows are masked only in the epilogue).
    int mA = m0 + lo;       if (mA >= M_TOT) mA = M_TOT - 1;
    int mB = m0 + 16 + lo;  if (mB >= M_TOT) mB = M_TOT - 1;
    const float* xbase = x + (size_t)b * (CIN * PLANE) + hi * (2 * PLANE);
    const float* xa = xbase + (mA / WOUT) * WIN + (mA % WOUT);
    const float* xb = xbase + (mB / WOUT) * WIN + (mB % WOUT);

    // Preload all A elements: clauses of immediate-offset loads (merge to b96).
    float aregA[2 * KSTEPS];
    float aregB[2 * KSTEPS];
#pragma unroll
    for (int s = 0; s < KSTEPS; ++s) {
        aregA[2 * s]     = xa[offA(s, 0)];
        aregA[2 * s + 1] = xa[offA(s, 1)];
        aregB[2 * s]     = xb[offA(s, 0)];
        aregB[2 * s + 1] = xb[offA(s, 1)];
    }

    v8f accA0 = {}, accA1 = {}, accA2 = {}, accA3 = {};
    v8f accB0 = {}, accB1 = {}, accB2 = {}, accB3 = {};
    const v2f* __restrict__ pB = (const v2f*)packedB;

    // Double-buffered B: fetch step s+1 while the 8 WMMAs of step s issue.
    v2f nb0 = pB[0 * 32 + lane];
    v2f nb1 = pB[1 * 32 + lane];
    v2f nb2 = pB[2 * 32 + lane];
    v2f nb3 = pB[3 * 32 + lane];

#pragma unroll
    for (int s = 0; s < KSTEPS; ++s) {
        v2f cb0 = nb0, cb1 = nb1, cb2 = nb2, cb3 = nb3;
        if (s + 1 < KSTEPS) {
            nb0 = pB[((s + 1) * 4 + 0) * 32 + lane];
            nb1 = pB[((s + 1) * 4 + 1) * 32 + lane];
            nb2 = pB[((s + 1) * 4 + 2) * 32 + lane];
            nb3 = pB[((s + 1) * 4 + 3) * 32 + lane];
        }
        v2f aA = {aregA[2 * s], aregA[2 * s + 1]};
        v2f aB = {aregB[2 * s], aregB[2 * s + 1]};
        accA0 = __builtin_amdgcn_wmma_f32_16x16x4_f32(false, aA, false, cb0,
                                                      (short)0, accA0, false, false);
        accB0 = __builtin_amdgcn_wmma_f32_16x16x4_f32(false, aB, false, cb0,
                                                      (short)0, accB0, false, false);
        accA1 = __builtin_amdgcn_wmma_f32_16x16x4_f32(false, aA, false, cb1,
                                                      (short)0, accA1, false, false);
        accB1 = __builtin_amdgcn_wmma_f32_16x16x4_f32(false, aB, false, cb1,
                                                      (short)0, accB1, false, false);
        accA2 = __builtin_amdgcn_wmma_f32_16x16x4_f32(false, aA, false, cb2,
                                                      (short)0, accA2, false, false);
        accB2 = __builtin_amdgcn_wmma_f32_16x16x4_f32(false, aB, false, cb2,
                                                      (short)0, accB2, false, false);
        accA3 = __builtin_amdgcn_wmma_f32_16x16x4_f32(false, aA, false, cb3,
                                                      (short)0, accA3, false, false);
        accB3 = __builtin_amdgcn_wmma_f32_16x16x4_f32(false, aB, false, cb3,
                                                      (short)0, accB3, false, false);
    }

    // Epilogue. C layout: N = lane%16 (lane-uniform channel),
    //           element j holds M = msub + 8*hi + j (consecutive).
    const int mbaseA = m0 + 8 * hi;
    const int mbaseB = m0 + 16 + 8 * hi;
    const int gbase  = lo >> 2;
    {
        const size_t p0 = ((size_t)b * COUT + (0 + lo)) * M_TOT;
        epilogue_tile(convout, p0, mbaseA, accA0, bias[0 + lo],  sgrp, 0 + gbase);
        epilogue_tile(convout, p0, mbaseB, accB0, bias[0 + lo],  sgrp, 0 + gbase);
        const size_t p1 = ((size_t)b * COUT + (16 + lo)) * M_TOT;
        epilogue_tile(convout, p1, mbaseA, accA1, bias[16 + lo], sgrp, 4 + gbase);
        epilogue_tile(convout, p1, mbaseB, accB1, bias[16 + lo], sgrp, 4 + gbase);
        const size_t p2 = ((size_t)b * COUT + (32 + lo)) * M_TOT;
        epilogue_tile(convout, p2, mbaseA, accA2, bias[32 + lo], sgrp, 8 + gbase);
        epilogue_tile(convout, p2, mbaseB, accB2, bias[32 + lo], sgrp, 8 + gbase);
        const size_t p3 = ((size_t)b * COUT + (48 + lo)) * M_TOT;
        epilogue_tile(convout, p3, mbaseA, accA3, bias[48 + lo], sgrp, 12 + gbase);
        epilogue_tile(convout, p3, mbaseB, accB3, bias[48 + lo], sgrp, 12 + gbase);
    }

    __syncthreads();
    if (tid < GROUPS * 2)
        atomicAdd(&stats[b * (GROUPS * 2) + tid], sgrp[tid]);
}

// ---------------------------------------------------------------------------
// Kernel 2: normalize (folded affine) + maxpool 4x4 + clamp
//   max(A*x+B) over window == A>0 ? A*max(x)+B : A*min(x)+B
// ---------------------------------------------------------------------------
__global__ __launch_bounds__(256) void finalize_kernel(
    const float* __restrict__ conv,     // [128, 64, 15876]
    const float* __restrict__ stats,    // [128, 16, 2]
    const float* __restrict__ gn_w,
    const float* __restrict__ gn_b,
    const float* __restrict__ scale,
    float* __restrict__ out)            // [128, 64, 31, 31]
{
    const int total = BATCH * COUT * POOLED * POOLED;
    int idx = blockIdx.x * 256 + threadIdx.x;
    if (idx >= total) return;

    int pw = idx % POOLED;
    int t  = idx / POOLED;
    int ph = t % POOLED;  t /= POOLED;
    int c  = t % COUT;
    int b  = t / COUT;
    int g  = c >> 2;

    const float invN = 1.0f / (float)(4 * M_TOT);
    float sum  = stats[b * (GROUPS * 2) + 2 * g];
    float sumq = stats[b * (GROUPS * 2) + 2 * g + 1];
    float mean = sum * invN;
    float var  = sumq * invN - mean * mean;
    float inv  = rsqrtf(var + GN_EPS);

    float A = gn_w[c] * inv * scale[c];
    float B = (gn_b[c] - mean * gn_w[c] * inv) * scale[c];

    const float* p = conv + ((size_t)b * COUT + c) * M_TOT
                          + (4 * ph) * WOUT + 4 * pw;
    float mx = -3.402823466e+38f;
    float mn =  3.402823466e+38f;
#pragma unroll
    for (int r = 0; r < 4; ++r) {
        // row and column offsets are both even -> 8B-aligned v2f
        const v2f* q = (const v2f*)(p + r * WOUT);
        v2f u = __builtin_nontemporal_load(q);
        v2f v = __builtin_nontemporal_load(q + 1);
        mx = fmaxf(mx, fmaxf(fmaxf(u.x, u.y), fmaxf(v.x, v.y)));
        mn = fminf(mn, fminf(fminf(u.x, u.y), fminf(v.x, v.y)));
    }
    float r = A * (A > 0.0f ? mx : mn) + B;
    r = fminf(fmaxf(r, 0.0f), 1.0f);
    out[idx] = r;
}

// ---------------------------------------------------------------------------
extern "C" void kernel_launch(void* const* d_in, const int* in_sizes, int n_in,
                              void* d_out, int out_size, void* d_ws, size_t ws_size,
                              hipStream_t stream)
{
    const float* x      = (const float*)d_in[0];
    const float* w      = (const float*)d_in[1];
    const float* bias   = (const float*)d_in[2];
    const float* gn_w   = (const float*)d_in[3];
    const float* gn_b   = (const float*)d_in[4];
    const float* scale  = (const float*)d_in[5];
    float* out = (float*)d_out;

    // ws layout: stats at 0 (16KB, zeroed each call), conv intermediate from 4KB
    float* stats = (float*)d_ws;
    float* conv  = (float*)((char*)d_ws + 4096);

    hipMemsetAsync(stats, 0, BATCH * GROUPS * 2 * sizeof(float), stream);

    dim3 g1((M_TOT + 255) / 256, BATCH);   // 63 x 128
    conv_stats_kernel<<<g1, 256, 0, stream>>>(x, w, bias, conv, stats);

    const int total = BATCH * COUT * POOLED * POOLED;
    finalize_kernel<<<(total + 255) / 256, 256, 0, stream>>>(
        conv, stats, gn_w, gn_b, scale, out);
}